// CtGmarlModel_19628000542918
// MI455X (gfx1250) — compile-verified
//
#include <hip/hip_runtime.h>
#include <hip/hip_bf16.h>
#include <math.h>

typedef __attribute__((ext_vector_type(2))) float v2f;
typedef __attribute__((ext_vector_type(8))) float v8f;
typedef int v4i __attribute__((vector_size(16)));

#define INROW 10517
#define AS1 __attribute__((address_space(1)))
#define AS3 __attribute__((address_space(3)))

#if defined(__has_builtin)
#if __has_builtin(__builtin_amdgcn_global_load_async_to_lds_b128) && \
    __has_builtin(__builtin_amdgcn_s_wait_asynccnt)
#define USE_ASYNC_LDS 1
#endif
#endif

// One 16x16 f32 output tile via V_WMMA_F32_16X16X4_F32, looping K in steps of 4.
// A: pointer to tile row 0, col 0 (row-major, leading dim lda)
// B: pointer to row 0 of this tile's 16 columns (row-major, leading dim ldb)
// Per ISA layouts: A lane L(li=L&15,half=L>>4): {A[li][2h], A[li][2h+1]};
//                  B: {B[2h][li], B[2h+1][li]}; D vgpr r -> M=r+8h, N=li.
__device__ __forceinline__ v8f wmma_k_f32(const float* __restrict__ A, int lda,
                                          const float* __restrict__ B, int ldb,
                                          int K, v8f c) {
  const int lane = threadIdx.x & 31;
  const int half = lane >> 4;
  const int li   = lane & 15;
  for (int k0 = 0; k0 < K; k0 += 4) {
    const int ka = k0 + 2 * half;
    v2f a, b;
    a.x = A[li * lda + ka];
    a.y = A[li * lda + ka + 1];
    b.x = B[ka * ldb + li];
    b.y = B[(ka + 1) * ldb + li];
    c = __builtin_amdgcn_wmma_f32_16x16x4_f32(false, a, false, b, (short)0, c,
                                              false, false);
  }
  return c;
}

// Generic GEMM: C[M,N] = act(A[M,K] @ B[K,N] + bias). 4 waves/block, 1 tile/wave.
// grid = (Ntiles/4, Mtiles), block = 128.
__global__ __launch_bounds__(128)
void gemm_wmma_bias_act(const float* __restrict__ A, int lda,
                        const float* __restrict__ B, int ldb,
                        const float* __restrict__ bias,
                        float* __restrict__ C, int ldc, int K, int relu) {
  const int wave = threadIdx.x >> 5;
  const int lane = threadIdx.x & 31;
  const int half = lane >> 4, li = lane & 15;
  const int mt = blockIdx.y;
  const int nt = blockIdx.x * 4 + wave;
  v8f c = {};
  c = wmma_k_f32(A + (size_t)(mt * 16) * lda, lda, B + nt * 16, ldb, K, c);
  const int col = nt * 16 + li;
  const float bv = bias ? bias[col] : 0.f;
#pragma unroll
  for (int r = 0; r < 8; ++r) {
    int row = mt * 16 + r + 8 * half;
    float v = c[r] + bv;
    if (relu) v = v > 0.f ? v : 0.f;
    C[(size_t)row * ldc + col] = v;
  }
}

// GAT collapse: only node 0 is nonzero, so alpha[:, :, i, 0] has a closed form.
// x_gat[s, h*32+d] = mean_i elu(alpha_i0[h] * Wh0[h,d]).
__global__ __launch_bounds__(128)
void gat_combine(const float* __restrict__ inputs,
                 const float* __restrict__ wh0,   // (2048,128) = Wh at node 0
                 const float* __restrict__ a_src, // (4,32) flat
                 const float* __restrict__ a_dst,
                 float* __restrict__ xgat) {      // (2048,128)
  __shared__ float s0s[4], d0s[4];
  __shared__ int cntS[100];
  __shared__ unsigned char ai0S[100];
  const int s = blockIdx.x;
  const int t = threadIdx.x;          // t = h*32+d; wave == head
  const int h = t >> 5;
  // streaming 82MB adjacency slab: hint the prefetcher (global_prefetch_b8)
  if (t < 100)
    __builtin_prefetch(inputs + (size_t)s * INROW + 132 + t * 100, 0, 1);
  const float whv = wh0[(size_t)s * 128 + t];
  float ps = whv * a_src[t];
  float pd = whv * a_dst[t];
#pragma unroll
  for (int m = 16; m >= 1; m >>= 1) {
    ps += __shfl_xor(ps, m, 32);
    pd += __shfl_xor(pd, m, 32);
  }
  if ((t & 31) == 0) { s0s[h] = ps; d0s[h] = pd; }
  if (t < 100) {  // row degree (incl. eye) and adj[i,0] flag
    const float* row = inputs + (size_t)s * INROW + 132 + t * 100;
    int cnt = 0;
    for (int j = 0; j < 100; ++j) cnt += ((row[j] > 0.f) || (j == t)) ? 1 : 0;
    cntS[t] = cnt;
    ai0S[t] = ((row[0] > 0.f) || (t == 0)) ? 1 : 0;
  }
  __syncthreads();
  const float s0 = s0s[h], d0 = d0s[h];
  auto leaky = [](float v) { return v > 0.f ? v : 0.2f * v; };
  const float e_s  = expf(leaky(s0));
  const float e_d  = expf(leaky(d0));
  const float e_sd = expf(leaky(s0 + d0));
  float accv = 0.f;
  {  // i == 0: e[0,0]=leaky(s0+d0), e[0,j>0 adj]=leaky(s0)
    float alpha = e_sd / (e_sd + (float)(cntS[0] - 1) * e_s);
    float v = alpha * whv;
    accv += v > 0.f ? v : (expf(v) - 1.f);
  }
  for (int i = 1; i < 100; ++i) {  // e[i,0]=leaky(d0), e[i,j>0 adj]=0
    float alpha = 0.f;
    if (ai0S[i]) alpha = e_d / (e_d + (float)(cntS[i] - 1));
    float v = alpha * whv;
    accv += v > 0.f ? v : (expf(v) - 1.f);
  }
  xgat[(size_t)s * 128 + t] = accv * 0.01f;  // mean over 100 nodes
}

// Persistent single-block RK4-ODE + GRU scan. State, temporaries and the ODE
// weights live in LDS (~208KB << 320KB/WGP). ODE weights are staged once via
// the gfx1250 async copy path (GLOBAL_LOAD_ASYNC_TO_LDS_B128 + s_wait_asynccnt)
// so the 8 GEMMs/step on the serial critical path hit LDS, not L2.
// 16 wave32s, one 16x16 WMMA tile each (2 Mtiles x 8 Ntiles).
__global__ __launch_bounds__(512)
void scan_kernel(const float* __restrict__ inputs, const float* __restrict__ h0,
                 const float* __restrict__ xgat,  // (B*T,128)
                 const float* __restrict__ ode_w1, const float* __restrict__ ode_b1,
                 const float* __restrict__ ode_w2, const float* __restrict__ ode_b2,
                 const float* __restrict__ gru_wi, const float* __restrict__ gru_wh,
                 const float* __restrict__ gru_bi, const float* __restrict__ gru_bh,
                 float* __restrict__ x_rnn, float* __restrict__ hT_out) {
  __shared__ float w1s[128 * 128];  // ode_w1 staged in LDS (64KB)
  __shared__ float w2s[128 * 128];  // ode_w2 staged in LDS (64KB)
  __shared__ float h[32 * 128];     // state; later reused for inn
  __shared__ float karg[32 * 128];  // f argument / output; later r-sum
  __shared__ float ktmp[32 * 128];  // f intermediate; later z-sum
  __shared__ float acc[32 * 128];   // RK4 accumulator -> h_ode
  __shared__ float gbuf[32 * 128];  // hn
  __shared__ float dts[32];
  const int tid  = threadIdx.x;
  const int wave = tid >> 5;
  const int lane = tid & 31;
  const int half = lane >> 4, li = lane & 15;
  const int mt = (wave >> 3) & 1;  // 0..1  (M = 32)
  const int nt = wave & 7;         // 0..7  (N = 128)

#ifdef USE_ASYNC_LDS
  {
    AS3 v4i* l1 = (AS3 v4i*)w1s;
    AS3 v4i* l2 = (AS3 v4i*)w2s;
    AS1 v4i* g1 = (AS1 v4i*)ode_w1;
    AS1 v4i* g2 = (AS1 v4i*)ode_w2;
    for (int idx = tid; idx < (128 * 128) / 4; idx += 512) {
      __builtin_amdgcn_global_load_async_to_lds_b128(g1 + idx, l1 + idx, 0, 0);
      __builtin_amdgcn_global_load_async_to_lds_b128(g2 + idx, l2 + idx, 0, 0);
    }
    __builtin_amdgcn_s_wait_asynccnt(0);
  }
#else
  for (int e = tid; e < 128 * 128; e += 512) {
    w1s[e] = ode_w1[e];
    w2s[e] = ode_w2[e];
  }
#endif
  for (int e = tid; e < 4096; e += 512) h[e] = h0[e];
  __syncthreads();

  for (int t = 0; t < 64; ++t) {
    if (tid < 32) {
      dts[tid] = inputs[(size_t)(tid * 64 + t) * INROW + 10132];
      if (t + 1 < 64)  // prefetch next step's GRU input rows
        __builtin_prefetch(xgat + (size_t)(tid * 64 + t + 1) * 128, 0, 1);
    }
    for (int e = tid; e < 4096; e += 512) {
      float hv = h[e];
      karg[e] = hv;
      acc[e]  = hv;
    }
    __syncthreads();

    const float ca[4] = {1.f / 6.f, 2.f / 6.f, 2.f / 6.f, 1.f / 6.f};
    const float aa[4] = {0.5f, 0.5f, 1.f, 0.f};
    for (int s = 0; s < 4; ++s) {
      {  // ktmp = tanh(karg @ ode_w1 + b1)
        v8f c = {};
        c = wmma_k_f32(karg + mt * 16 * 128, 128, w1s + nt * 16, 128, 128, c);
        int col = nt * 16 + li;
        float bv = ode_b1[col];
#pragma unroll
        for (int r = 0; r < 8; ++r)
          ktmp[(mt * 16 + r + 8 * half) * 128 + col] = tanhf(c[r] + bv);
      }
      __syncthreads();
      {  // karg = tanh(ktmp @ ode_w2 + b2)   (in-place f output)
        v8f c = {};
        c = wmma_k_f32(ktmp + mt * 16 * 128, 128, w2s + nt * 16, 128, 128, c);
        int col = nt * 16 + li;
        float bv = ode_b2[col];
#pragma unroll
        for (int r = 0; r < 8; ++r)
          karg[(mt * 16 + r + 8 * half) * 128 + col] = tanhf(c[r] + bv);
      }
      __syncthreads();
      for (int e = tid; e < 4096; e += 512) {  // RK4 accumulate + next arg
        float dt = dts[e >> 7];
        float ks = karg[e];
        acc[e] += dt * ca[s] * ks;
        if (s < 3) karg[e] = h[e] + dt * aa[s] * ks;
      }
      __syncthreads();
    }
    // acc == h_ode. GRU gates: gh = h_ode @ gru_wh + bh (3 chunks of 128 cols)
#pragma unroll
    for (int ch = 0; ch < 3; ++ch) {
      v8f c = {};
      c = wmma_k_f32(acc + mt * 16 * 128, 128, gru_wh + ch * 128 + nt * 16, 384,
                     128, c);
      int col = nt * 16 + li;
      float bv = gru_bh[ch * 128 + col];
      float* dst = (ch == 0) ? karg : (ch == 1) ? ktmp : gbuf;  // hr, hz, hn
#pragma unroll
      for (int r = 0; r < 8; ++r)
        dst[(mt * 16 + r + 8 * half) * 128 + col] = c[r] + bv;
    }
    __syncthreads();
    // gi = x_t @ gru_wi + bi, A rows strided through (B,T,128) global buffer
#pragma unroll
    for (int ch = 0; ch < 3; ++ch) {
      v8f c = {};
      c = wmma_k_f32(xgat + t * 128 + (size_t)(mt * 16) * (64 * 128), 64 * 128,
                     gru_wi + ch * 128 + nt * 16, 384, 128, c);
      int col = nt * 16 + li;
      float bv = gru_bi[ch * 128 + col];
#pragma unroll
      for (int r = 0; r < 8; ++r) {
        int e = (mt * 16 + r + 8 * half) * 128 + col;
        if (ch == 0)      karg[e] += c[r] + bv;  // ir+hr
        else if (ch == 1) ktmp[e] += c[r] + bv;  // iz+hz
        else              h[e]     = c[r] + bv;  // inn (old h is dead)
      }
    }
    __syncthreads();
    for (int e = tid; e < 4096; e += 512) {  // GRU update
      int b = e >> 7, j = e & 127;
      float r = 1.f / (1.f + expf(-karg[e]));
      float z = 1.f / (1.f + expf(-ktmp[e]));
      float n = tanhf(h[e] + r * gbuf[e]);
      float hnew = (1.f - z) * n + z * acc[e];
      h[e] = hnew;
      x_rnn[(size_t)(b * 64 + t) * 128 + j] = hnew;
    }
    __syncthreads();
  }
  for (int e = tid; e < 4096; e += 512) hT_out[e] = h[e];
}

// logits = x_rnn @ act_w + act_b (masked), value = x_rnn @ val_w + val_b.
__global__ __launch_bounds__(160)
void heads_kernel(const float* __restrict__ inputs, const float* __restrict__ x_rnn,
                  const float* __restrict__ act_w, const float* __restrict__ act_b,
                  const float* __restrict__ val_w, const float* __restrict__ val_b,
                  float* __restrict__ out_logits, float* __restrict__ out_value) {
  __shared__ float row[128];
  const int s = blockIdx.x;
  const int t = threadIdx.x;
  if (t < 128) row[t] = x_rnn[(size_t)s * 128 + t];
  __syncthreads();
  if (t < 132) {
    float accv = act_b[t];
    for (int k = 0; k < 128; ++k) accv += row[k] * act_w[k * 132 + t];
    float mask = inputs[(size_t)s * INROW + t];
    out_logits[(size_t)s * 132 + t] = (mask == 0.f) ? -1e10f : accv;
  } else if (t == 132) {
    float accv = val_b[0];
    for (int k = 0; k < 128; ++k) accv += row[k] * val_w[k];
    out_value[s] = accv;
  }
}

extern "C" void kernel_launch(void* const* d_in, const int* in_sizes, int n_in,
                              void* d_out, int out_size, void* d_ws, size_t ws_size,
                              hipStream_t stream) {
  const float* inputs = (const float*)d_in[0];
  const float* h0     = (const float*)d_in[1];
  const float* fc_w   = (const float*)d_in[2];
  const float* fc_b   = (const float*)d_in[3];
  const float* gat_w  = (const float*)d_in[4];
  const float* a_src  = (const float*)d_in[5];
  const float* a_dst  = (const float*)d_in[6];
  const float* ode_w1 = (const float*)d_in[7];
  const float* ode_b1 = (const float*)d_in[8];
  const float* ode_w2 = (const float*)d_in[9];
  const float* ode_b2 = (const float*)d_in[10];
  const float* gru_wi = (const float*)d_in[11];
  const float* gru_wh = (const float*)d_in[12];
  const float* gru_bi = (const float*)d_in[13];
  const float* gru_bh = (const float*)d_in[14];
  const float* act_w  = (const float*)d_in[15];
  const float* act_b  = (const float*)d_in[16];
  const float* val_w  = (const float*)d_in[17];
  const float* val_b  = (const float*)d_in[18];

  float* ws       = (float*)d_ws;
  float* buf_x    = ws;                     // 2048*256
  float* buf_wh0  = buf_x + 2048 * 256;     // 2048*128
  float* buf_xgat = buf_wh0 + 2048 * 128;   // 2048*128
  float* x_rnn    = buf_xgat + 2048 * 128;  // 2048*128

  float* out        = (float*)d_out;
  float* out_logits = out;                  // 2048*132
  float* out_hT     = out + 2048 * 132;     // 32*128
  float* out_val    = out_hT + 32 * 128;    // 2048

  // FC: relu([obs|siem](2048x384) @ fc_w(384x256) + fc_b)
  gemm_wmma_bias_act<<<dim3(16 / 4, 128), 128, 0, stream>>>(
      inputs + 10133, INROW, fc_w, 256, fc_b, buf_x, 256, 384, 1);
  // Wh at node 0: buf_x(2048x256) @ gat_w(256x128)
  gemm_wmma_bias_act<<<dim3(8 / 4, 128), 128, 0, stream>>>(
      buf_x, 256, gat_w, 128, nullptr, buf_wh0, 128, 256, 0);
  gat_combine<<<2048, 128, 0, stream>>>(inputs, buf_wh0, a_src, a_dst, buf_xgat);
  scan_kernel<<<1, 512, 0, stream>>>(inputs, h0, buf_xgat, ode_w1, ode_b1, ode_w2,
                                     ode_b2, gru_wi, gru_wh, gru_bi, gru_bh, x_rnn,
                                     out_hT);
  heads_kernel<<<2048, 160, 0, stream>>>(inputs, x_rnn, act_w, act_b, val_w, val_b,
                                         out_logits, out_val);
}